// GCN_47416438948092
// MI455X (gfx1250) — compile-verified
//
#include <hip/hip_runtime.h>
#include <hip/hip_bf16.h>
#include <math.h>

typedef __attribute__((ext_vector_type(2))) float v2f;
typedef __attribute__((ext_vector_type(8))) float v8f;

#define N_NODES 50000
#define N_EDGES 800000
#define IN_DIM  128
#define HID_DIM 96
#define OUT_DIM 40
#define OUT_PAD 48

__global__ void zero_f32(float* __restrict__ p, int n) {
    int i = blockIdx.x * blockDim.x + threadIdx.x;
    if (i < n) p[i] = 0.0f;
}

// -------- GEMM1: H1[50000x96] = x[50000x128] @ W1[128x96] via fp32 WMMA ----
// One block per 16-row tile (50000 = 3125*16 exactly). 6 waves = 6 N-tiles.
__global__ __launch_bounds__(192) void gemm1_wmma(const float* __restrict__ x,
                                                  const float* __restrict__ W1,
                                                  float* __restrict__ H1) {
    __shared__ float w1s[IN_DIM * HID_DIM]; // 48 KB
    __shared__ float xs[16 * IN_DIM];       // 8 KB
    const int tid = threadIdx.x;
    const int m0  = blockIdx.x * 16;
    for (int i = tid; i < IN_DIM * HID_DIM; i += 192) w1s[i] = W1[i];
    for (int i = tid; i < 16 * IN_DIM; i += 192)      xs[i]  = x[(size_t)m0 * IN_DIM + i];
    __syncthreads();

    const int wave = tid >> 5;
    const int lane = tid & 31;
    const int hi   = lane >> 4;  // 0: lanes 0-15, 1: lanes 16-31
    const int l15  = lane & 15;
    const int n0   = wave * 16;

    v8f acc = {};
    for (int k = 0; k < IN_DIM; k += 4) {
        const int ka = k + 2 * hi;
        v2f a, b;
        // A 16x4 f32: lanes 0-15 hold K=0,1; lanes 16-31 hold K=2,3 (M = lane&15)
        a.x = xs[l15 * IN_DIM + ka];
        a.y = xs[l15 * IN_DIM + ka + 1];
        // B 4x16 f32: VGPR v, lanes 0-15 -> K=v, lanes 16-31 -> K=v+2 (N = n0+(lane&15))
        b.x = w1s[(ka)     * HID_DIM + n0 + l15];
        b.y = w1s[(ka + 1) * HID_DIM + n0 + l15];
        acc = __builtin_amdgcn_wmma_f32_16x16x4_f32(false, a, false, b,
                                                    (short)0, acc, false, false);
    }
    // C/D 16x16 f32: VGPR r -> M = r + 8*hi, N = n0 + (lane&15)
#pragma unroll
    for (int r = 0; r < 8; ++r)
        H1[(size_t)(m0 + r + 8 * hi) * HID_DIM + n0 + l15] = acc[r];
}

// -------- GEMM2: H2[50000x40] = h[50000x96] @ W2[96x40], N padded to 48 -----
__global__ __launch_bounds__(96) void gemm2_wmma(const float* __restrict__ h,
                                                 const float* __restrict__ W2,
                                                 float* __restrict__ H2) {
    __shared__ float w2s[HID_DIM * OUT_PAD]; // 18 KB (zero-padded cols 40..47)
    __shared__ float hs[16 * HID_DIM];       // 6 KB
    const int tid = threadIdx.x;
    const int m0  = blockIdx.x * 16;
    for (int i = tid; i < HID_DIM * OUT_PAD; i += 96) {
        int r = i / OUT_PAD, c = i % OUT_PAD;
        w2s[i] = (c < OUT_DIM) ? W2[r * OUT_DIM + c] : 0.0f;
    }
    for (int i = tid; i < 16 * HID_DIM; i += 96) hs[i] = h[(size_t)m0 * HID_DIM + i];
    __syncthreads();

    const int wave = tid >> 5;
    const int lane = tid & 31;
    const int hi   = lane >> 4;
    const int l15  = lane & 15;
    const int n0   = wave * 16;

    v8f acc = {};
    for (int k = 0; k < HID_DIM; k += 4) {
        const int ka = k + 2 * hi;
        v2f a, b;
        a.x = hs[l15 * HID_DIM + ka];
        a.y = hs[l15 * HID_DIM + ka + 1];
        b.x = w2s[(ka)     * OUT_PAD + n0 + l15];
        b.y = w2s[(ka + 1) * OUT_PAD + n0 + l15];
        acc = __builtin_amdgcn_wmma_f32_16x16x4_f32(false, a, false, b,
                                                    (short)0, acc, false, false);
    }
    const int col = n0 + l15;
    if (col < OUT_DIM) {   // divergence only after the MMA loop
#pragma unroll
        for (int r = 0; r < 8; ++r)
            H2[(size_t)(m0 + r + 8 * hi) * OUT_DIM + col] = acc[r];
    }
}

// -------- COO SpMM scatter: out[dst] += val * sup[src], D features ----------
template <int D>
__global__ void spmm_atomic(const int* __restrict__ src, const int* __restrict__ dst,
                            const float* __restrict__ val, const float* __restrict__ sup,
                            float* __restrict__ out) {
    constexpr int D4 = D / 4;
    unsigned t = blockIdx.x * blockDim.x + threadIdx.x;
    if (t >= (unsigned)N_EDGES * D4) return;
    const int e = t / D4;
    const int f = (t - e * D4) * 4;
    const int s = src[e], d = dst[e];
    const float v = val[e];
    const float4 m = *reinterpret_cast<const float4*>(sup + (size_t)s * D + f);
    float* o = out + (size_t)d * D + f;
    atomicAdd(o + 0, v * m.x);
    atomicAdd(o + 1, v * m.y);
    atomicAdd(o + 2, v * m.z);
    atomicAdd(o + 3, v * m.w);
}

__global__ void bias_relu(float* __restrict__ hbuf, const float* __restrict__ b1) {
    unsigned i = blockIdx.x * blockDim.x + threadIdx.x;
    if (i >= (unsigned)N_NODES * HID_DIM) return;
    float t = hbuf[i] + b1[i % HID_DIM];
    hbuf[i] = t > 0.0f ? t : 0.0f;
}

__global__ void finalize_lsm(float* __restrict__ out, const float* __restrict__ b2) {
    int i = blockIdx.x * blockDim.x + threadIdx.x;
    if (i >= N_NODES) return;
    float* row = out + (size_t)i * OUT_DIM;
    float r[OUT_DIM];
    float mx = -INFINITY;
#pragma unroll
    for (int j = 0; j < OUT_DIM; ++j) { r[j] = row[j] + b2[j]; mx = fmaxf(mx, r[j]); }
    float s = 0.0f;
#pragma unroll
    for (int j = 0; j < OUT_DIM; ++j) s += __expf(r[j] - mx);
    const float ls = __logf(s) + mx;
#pragma unroll
    for (int j = 0; j < OUT_DIM; ++j) row[j] = r[j] - ls;
}

extern "C" void kernel_launch(void* const* d_in, const int* in_sizes, int n_in,
                              void* d_out, int out_size, void* d_ws, size_t ws_size,
                              hipStream_t stream) {
    const float* x  = (const float*)d_in[0];
    const float* ev = (const float*)d_in[1];
    const float* W1 = (const float*)d_in[2];
    const float* b1 = (const float*)d_in[3];
    const float* W2 = (const float*)d_in[4];
    const float* b2 = (const float*)d_in[5];
    const int*   es = (const int*)d_in[6];
    const int*   ed = (const int*)d_in[7];
    float* out = (float*)d_out;

    float* H1   = (float*)d_ws;                     // 50000*96 fp32 (19.2 MB)
    float* Hagg = H1   + (size_t)N_NODES * HID_DIM; // 50000*96 fp32 (19.2 MB)
    float* H2   = Hagg + (size_t)N_NODES * HID_DIM; // 50000*40 fp32 (8 MB)

    // Zero the two scatter accumulators (kernel-based: graph-capture safe).
    {
        int n = N_NODES * HID_DIM;
        zero_f32<<<(n + 255) / 256, 256, 0, stream>>>(Hagg, n);
        n = N_NODES * OUT_DIM;
        zero_f32<<<(n + 255) / 256, 256, 0, stream>>>(out, n);
    }

    // Layer 1
    gemm1_wmma<<<N_NODES / 16, 192, 0, stream>>>(x, W1, H1);
    {
        unsigned total = (unsigned)N_EDGES * (HID_DIM / 4);
        spmm_atomic<HID_DIM><<<(total + 255) / 256, 256, 0, stream>>>(es, ed, ev, H1, Hagg);
    }
    {
        unsigned n = (unsigned)N_NODES * HID_DIM;
        bias_relu<<<(n + 255) / 256, 256, 0, stream>>>(Hagg, b1);
    }

    // Layer 2
    gemm2_wmma<<<N_NODES / 16, 96, 0, stream>>>(Hagg, W2, H2);
    {
        unsigned total = (unsigned)N_EDGES * (OUT_DIM / 4);
        spmm_atomic<OUT_DIM><<<(total + 255) / 256, 256, 0, stream>>>(es, ed, ev, H2, out);
    }

    finalize_lsm<<<(N_NODES + 255) / 256, 256, 0, stream>>>(out, b2);
}